// MultimodalBinder_23270132810220
// MI455X (gfx1250) — compile-verified
//
#include <hip/hip_runtime.h>
#include <hip/hip_bf16.h>
#include <math.h>

// ---------------------------------------------------------------------------
// MultimodalBinder on MI455X (gfx1250, wave32, WMMA)
//
// Pipeline:
//   1) gemm_enc  : encoded = x @ Wf^T + bf  (B x 256)  and
//                  phases  = (x @ Wp^T + bp) mod 2pi   (B x 64 per modality)
//                  -> fp32 WMMA 16x16x4, X tile staged in LDS
//   2) kuramoto  : persistent kernel, 10 steps. Per step:
//                  forces = (sin(ph)*(C@K) - cos(ph)*(S@K))/N  via two
//                  192x192 fp32-WMMA GEMMs per 16-row wave tile.
//                  K (147KB), phases/cos/sin wave tiles all live in the
//                  320KB LDS for the whole kernel.
//   3) rowstats  : per-row mean cos / mean sin -> phase_mod, R
//   4) fusion    : bound = (E * (0.5+0.5*pm)) @ W_fusion^T + b  (WMMA)
//   5) finalize  : binding matrix (3x3), total_binding broadcast
// ---------------------------------------------------------------------------

#define BATCH   2048
#define NOSC    64
#define NTOT    192           // 3 * NOSC
#define BDIM    256
#define FDIM    768           // 3 * BDIM
#define DT_C    0.01f
#define NSTEPS  10
#define TWO_PI_F 6.28318530717958647692f

typedef __attribute__((ext_vector_type(2))) float v2f;
typedef __attribute__((ext_vector_type(8))) float v8f;

__device__ __forceinline__ v8f wmma_f32_16x16x4(v2f a, v2f b, v8f c) {
    // 8-arg pattern: (neg_a, A, neg_b, B, c_mod, C, reuse_a, reuse_b)
    return __builtin_amdgcn_wmma_f32_16x16x4_f32(
        false, a, false, b, (short)0, c, false, false);
}

__device__ __forceinline__ float wrap2pi(float v) {
    float r = fmodf(v, TWO_PI_F);
    if (r < 0.0f) r += TWO_PI_F;
    return r;
}

// ---------------------------------------------------------------------------
// Generic encoder GEMM: out[m, colOff+n] = sum_k X[m,k]*W[n,k] + bias[n]
// grid = (BATCH/16, N/64), block = 128 (4 waves, one 16x16 n-tile per wave)
// ---------------------------------------------------------------------------
__global__ __launch_bounds__(128)
void gemm_enc(const float* __restrict__ X, const float* __restrict__ W,
              const float* __restrict__ bias, float* __restrict__ out,
              int Kd, int ldOut, int colOff, int doMod) {
    __shared__ float xs[16 * 64];
    const int tid  = threadIdx.x;
    const int lane = tid & 31;
    const int wave = tid >> 5;
    const int m0   = blockIdx.x * 16;
    const int n0   = blockIdx.y * 64 + wave * 16;
    const int c    = lane & 15;      // A row / B,D column within tile
    const int h    = lane >> 4;      // lane half

    v8f acc = {};
    const float* Wrow = W + (size_t)(n0 + c) * Kd;

    for (int kc = 0; kc < Kd; kc += 64) {
        __syncthreads();
        // stage X[m0:m0+16, kc:kc+64] (coalesced over columns)
        for (int i = tid; i < 16 * 64; i += 128) {
            int r = i >> 6, cc = i & 63;
            xs[i] = X[(size_t)(m0 + r) * Kd + kc + cc];
        }
        __syncthreads();
        for (int kk = 0; kk < 64; kk += 4) {
            v2f a, b;
            const int ko = kk + 2 * h;
            a.x = xs[c * 64 + ko];       // A[l&15, 2h+0]
            a.y = xs[c * 64 + ko + 1];   // A[l&15, 2h+1]
            b.x = Wrow[kc + ko];         // B[2h+0, l&15] = W[n, k]
            b.y = Wrow[kc + ko + 1];     // B[2h+1, l&15]
            acc = wmma_f32_16x16x4(a, b, acc);
        }
    }

    const float bv = bias[n0 + c];
    for (int r = 0; r < 8; ++r) {
        int m = r + 8 * h;
        float v = acc[r] + bv;
        if (doMod) v = wrap2pi(v);
        out[(size_t)(m0 + m) * ldOut + colOff + n0 + c] = v;
    }
}

// ---------------------------------------------------------------------------
// Persistent Kuramoto kernel.
// grid = 32 blocks x 128 threads (4 waves); each wave owns 16 batch rows.
// LDS: K_sym (192x192) + omega (192) + per-wave ph/cos/sin (16x192 each).
// ---------------------------------------------------------------------------
__global__ __launch_bounds__(128)
void kuramoto(const float* __restrict__ IC,
              const float* __restrict__ oa, const float* __restrict__ ot,
              const float* __restrict__ ov,
              const float* __restrict__ Pin, float* __restrict__ Pout) {
    extern __shared__ float smem[];
    float* Ks = smem;                       // 192*192
    float* om = Ks + NTOT * NTOT;           // 192
    float* phB = om + NTOT;                 // 4 * 16*192
    float* csB = phB + 4 * 16 * NTOT;       // 4 * 16*192
    float* snB = csB + 4 * 16 * NTOT;       // 4 * 16*192

    const int tid  = threadIdx.x;
    const int lane = tid & 31;
    const int wave = tid >> 5;
    const int c    = lane & 15;
    const int h    = lane >> 4;

    // K = (IC + IC^T) * 0.5, built once per block
    for (int i = tid; i < NTOT * NTOT; i += 128) {
        int r = i / NTOT, cc = i - r * NTOT;
        Ks[i] = 0.5f * (IC[i] + IC[cc * NTOT + r]);
    }
    for (int i = tid; i < NTOT; i += 128)
        om[i] = (i < 64) ? oa[i] : ((i < 128) ? ot[i - 64] : ov[i - 128]);

    const int mw = blockIdx.x * 64 + wave * 16;   // first batch row of wave
    float* phw = phB + wave * 16 * NTOT;
    float* csw = csB + wave * 16 * NTOT;
    float* snw = snB + wave * 16 * NTOT;

    for (int i = lane; i < 16 * NTOT; i += 32)
        phw[i] = Pin[(size_t)mw * NTOT + i];      // rows are contiguous
    __syncthreads();

    const float invN = 1.0f / (float)NTOT;
    for (int step = 0; step < NSTEPS; ++step) {
        // sincos pass
        for (int i = lane; i < 16 * NTOT; i += 32) {
            float s, cc2;
            sincosf(phw[i], &s, &cc2);
            snw[i] = s; csw[i] = cc2;
        }
        __syncthreads();
        // two GEMMs (C@K, S@K) fused per 16-wide n-tile
        for (int nt = 0; nt < NTOT / 16; ++nt) {
            const int n0 = nt * 16;
            v8f accC = {}, accS = {};
            for (int k0 = 0; k0 < NTOT; k0 += 4) {
                const int kr = k0 + 2 * h;
                v2f b, aC, aS;
                b.x = Ks[(kr)     * NTOT + n0 + c];
                b.y = Ks[(kr + 1) * NTOT + n0 + c];
                const int ab = c * NTOT + kr;
                aC.x = csw[ab]; aC.y = csw[ab + 1];
                aS.x = snw[ab]; aS.y = snw[ab + 1];
                accC = wmma_f32_16x16x4(aC, b, accC);
                accS = wmma_f32_16x16x4(aS, b, accS);
            }
            // epilogue: integrate phases for this 16x16 tile
            for (int r = 0; r < 8; ++r) {
                int m = r + 8 * h;
                int idx = m * NTOT + n0 + c;
                float force = (snw[idx] * accC[r] - csw[idx] * accS[r]) * invN;
                float pn = phw[idx] + DT_C * (om[n0 + c] + force);
                phw[idx] = wrap2pi(pn);
            }
        }
        __syncthreads();
    }

    for (int i = lane; i < 16 * NTOT; i += 32)
        Pout[(size_t)mw * NTOT + i] = phw[i];
}

// ---------------------------------------------------------------------------
// Per-row circular stats: pm[b,m] = mean cos, R[b,m] = resultant length.
// grid = BATCH blocks x 192 threads.
// ---------------------------------------------------------------------------
__global__ __launch_bounds__(192)
void rowstats(const float* __restrict__ Ph, float* __restrict__ pm,
              float* __restrict__ Rv) {
    __shared__ float cb[NTOT], sb[NTOT];
    const int b = blockIdx.x, t = threadIdx.x;
    float s, cc;
    sincosf(Ph[(size_t)b * NTOT + t], &s, &cc);
    cb[t] = cc; sb[t] = s;
    __syncthreads();
    if (t < 3) {
        float mc = 0.0f, ms = 0.0f;
        for (int i = 0; i < NOSC; ++i) {
            mc += cb[t * NOSC + i];
            ms += sb[t * NOSC + i];
        }
        mc *= (1.0f / NOSC); ms *= (1.0f / NOSC);
        pm[b * 3 + t] = mc;
        Rv[b * 3 + t] = sqrtf(mc * mc + ms * ms);
    }
}

// ---------------------------------------------------------------------------
// Fusion GEMM: bound = (E * (0.5 + 0.5*pm[modality])) @ W_fusion^T + b_fusion
// grid = (BATCH/16, 256/64), block = 128.
// ---------------------------------------------------------------------------
__global__ __launch_bounds__(128)
void fusion_gemm(const float* __restrict__ E, const float* __restrict__ pm,
                 const float* __restrict__ Wf, const float* __restrict__ bias,
                 float* __restrict__ out) {
    __shared__ float xs[16 * 64];
    const int tid  = threadIdx.x;
    const int lane = tid & 31;
    const int wave = tid >> 5;
    const int m0   = blockIdx.x * 16;
    const int n0   = blockIdx.y * 64 + wave * 16;
    const int c    = lane & 15;
    const int h    = lane >> 4;

    v8f acc = {};
    const float* Wrow = Wf + (size_t)(n0 + c) * FDIM;

    for (int kc = 0; kc < FDIM; kc += 64) {
        __syncthreads();
        for (int i = tid; i < 16 * 64; i += 128) {
            int r = i >> 6, cc = i & 63;
            int kg = kc + cc;                       // global k index
            float scale = 0.5f + 0.5f * pm[(m0 + r) * 3 + (kg >> 8)];
            xs[i] = E[(size_t)(m0 + r) * FDIM + kg] * scale;
        }
        __syncthreads();
        for (int kk = 0; kk < 64; kk += 4) {
            const int ko = kk + 2 * h;
            v2f a, b;
            a.x = xs[c * 64 + ko];
            a.y = xs[c * 64 + ko + 1];
            b.x = Wrow[kc + ko];
            b.y = Wrow[kc + ko + 1];
            acc = wmma_f32_16x16x4(a, b, acc);
        }
    }

    const float bv = bias[n0 + c];
    for (int r = 0; r < 8; ++r) {
        int m = r + 8 * h;
        out[(size_t)(m0 + m) * BDIM + n0 + c] = acc[r] + bv;
    }
}

// ---------------------------------------------------------------------------
// Binding matrix + total_binding. 1 block x 256 threads.
// ---------------------------------------------------------------------------
__global__ __launch_bounds__(256)
void finalize(const float* __restrict__ Rv, float* __restrict__ out_bm,
              float* __restrict__ out_tb) {
    __shared__ float red[3 * 256];
    __shared__ float tot;
    const int t = threadIdx.x;
    float s01 = 0.0f, s02 = 0.0f, s12 = 0.0f;
    for (int b = t; b < BATCH; b += 256) {
        float r0 = Rv[b * 3 + 0], r1 = Rv[b * 3 + 1], r2 = Rv[b * 3 + 2];
        s01 += r0 * r1; s02 += r0 * r2; s12 += r1 * r2;
    }
    red[t] = s01; red[256 + t] = s02; red[512 + t] = s12;
    __syncthreads();
    for (int off = 128; off > 0; off >>= 1) {
        if (t < off) {
            red[t]       += red[t + off];
            red[256 + t] += red[256 + t + off];
            red[512 + t] += red[512 + t + off];
        }
        __syncthreads();
    }
    if (t == 0) {
        float b01 = red[0]   * (1.0f / BATCH);
        float b02 = red[256] * (1.0f / BATCH);
        float b12 = red[512] * (1.0f / BATCH);
        out_bm[0] = 1.0f; out_bm[1] = b01; out_bm[2] = b02;
        out_bm[3] = b01;  out_bm[4] = 1.0f; out_bm[5] = b12;
        out_bm[6] = b02;  out_bm[7] = b12;  out_bm[8] = 1.0f;
        tot = (b01 + b02 + b12) * (1.0f / 3.0f);
    }
    __syncthreads();
    const float tv = tot;
    for (int i = t; i < BATCH; i += 256) out_tb[i] = tv;
}

// ---------------------------------------------------------------------------
extern "C" void kernel_launch(void* const* d_in, const int* in_sizes, int n_in,
                              void* d_out, int out_size, void* d_ws, size_t ws_size,
                              hipStream_t stream) {
    const float* x_a  = (const float*)d_in[0];
    const float* Wf_a = (const float*)d_in[1];
    const float* bf_a = (const float*)d_in[2];
    const float* Wp_a = (const float*)d_in[3];
    const float* bp_a = (const float*)d_in[4];
    const float* x_t  = (const float*)d_in[5];
    const float* Wf_t = (const float*)d_in[6];
    const float* bf_t = (const float*)d_in[7];
    const float* Wp_t = (const float*)d_in[8];
    const float* bp_t = (const float*)d_in[9];
    const float* x_v  = (const float*)d_in[10];
    const float* Wf_v = (const float*)d_in[11];
    const float* bf_v = (const float*)d_in[12];
    const float* Wp_v = (const float*)d_in[13];
    const float* bp_v = (const float*)d_in[14];
    const float* om_a = (const float*)d_in[15];
    const float* om_t = (const float*)d_in[16];
    const float* om_v = (const float*)d_in[17];
    const float* IC   = (const float*)d_in[18];
    const float* W_fu = (const float*)d_in[19];
    const float* b_fu = (const float*)d_in[20];

    float* out = (float*)d_out;
    // output layout: bound (2048*256) | binding_matrix (9) | phases (2048*192)
    //                | total_binding (2048)
    float* outBound = out;
    float* outBm    = out + (size_t)BATCH * BDIM;
    float* outPh    = outBm + 9;
    float* outTb    = outPh + (size_t)BATCH * NTOT;

    // workspace layout (floats): E (2048*768) | P (2048*192) | pm | R
    float* E  = (float*)d_ws;
    float* P  = E + (size_t)BATCH * FDIM;
    float* pm = P + (size_t)BATCH * NTOT;
    float* Rv = pm + (size_t)BATCH * 3;

    // 1) encoders: encoded -> E (col blocks per modality), phases -> P
    gemm_enc<<<dim3(BATCH / 16, 4), 128, 0, stream>>>(x_a, Wf_a, bf_a, E,  512, FDIM,   0, 0);
    gemm_enc<<<dim3(BATCH / 16, 1), 128, 0, stream>>>(x_a, Wp_a, bp_a, P,  512, NTOT,   0, 1);
    gemm_enc<<<dim3(BATCH / 16, 4), 128, 0, stream>>>(x_t, Wf_t, bf_t, E,  768, FDIM, 256, 0);
    gemm_enc<<<dim3(BATCH / 16, 1), 128, 0, stream>>>(x_t, Wp_t, bp_t, P,  768, NTOT,  64, 1);
    gemm_enc<<<dim3(BATCH / 16, 4), 128, 0, stream>>>(x_v, Wf_v, bf_v, E, 1024, FDIM, 512, 0);
    gemm_enc<<<dim3(BATCH / 16, 1), 128, 0, stream>>>(x_v, Wp_v, bp_v, P, 1024, NTOT, 128, 1);

    // 2) persistent Kuramoto: K + omega + 3 per-wave tiles in LDS (~296 KB)
    const size_t smemBytes =
        (size_t)(NTOT * NTOT + NTOT + 3 * 4 * 16 * NTOT) * sizeof(float);
    kuramoto<<<BATCH / 64, 128, smemBytes, stream>>>(IC, om_a, om_t, om_v, P, outPh);

    // 3) circular statistics per row
    rowstats<<<BATCH, NTOT, 0, stream>>>(outPh, pm, Rv);

    // 4) fusion GEMM -> bound
    fusion_gemm<<<dim3(BATCH / 16, 4), 128, 0, stream>>>(E, pm, W_fu, b_fu, outBound);

    // 5) binding matrix + total_binding
    finalize<<<1, 256, 0, stream>>>(Rv, outBm, outTb);
}